// BioSleepXContrastive_90237262889286
// MI455X (gfx1250) — compile-verified
//
#include <hip/hip_runtime.h>
#include <hip/hip_bf16.h>

typedef __attribute__((ext_vector_type(2))) float v2f;
typedef __attribute__((ext_vector_type(8))) float v8f;

#if __has_builtin(__builtin_amdgcn_global_load_async_to_lds_b32) && \
    __has_builtin(__builtin_amdgcn_s_wait_asynccnt)
#define USE_ASYNC_LDS 1
typedef __attribute__((address_space(1))) int async_gint;
typedef __attribute__((address_space(3))) int async_lint;
#endif

// ---------------------------------------------------------------------------
// LayerNorm over rows of 80. mode 0: custom_ln (unbiased var, eps=1e-6 added
// to std). mode 1: std_ln (biased var, eps=1e-5 inside sqrt) followed by ReLU.
// ---------------------------------------------------------------------------
__global__ __launch_bounds__(128)
void ln80_kernel(const float* __restrict__ X, const float* __restrict__ A,
                 const float* __restrict__ Bp, float* __restrict__ Y, int mode)
{
    const int row = blockIdx.x;
    const int t = threadIdx.x;
    __shared__ float buf[128];
    __shared__ float stat[2];
    const float* xr = X + (size_t)row * 80;
    float v = (t < 80) ? xr[t] : 0.f;
    buf[t] = v;
    __syncthreads();
    for (int s = 64; s > 0; s >>= 1) { if (t < s) buf[t] += buf[t + s]; __syncthreads(); }
    if (t == 0) stat[0] = buf[0] * (1.f / 80.f);
    __syncthreads();
    const float mean = stat[0];
    const float dv = (t < 80) ? (v - mean) : 0.f;
    buf[t] = dv * dv;
    __syncthreads();
    for (int s = 64; s > 0; s >>= 1) { if (t < s) buf[t] += buf[t + s]; __syncthreads(); }
    if (t == 0) stat[1] = buf[0];
    __syncthreads();
    const float ss = stat[1];
    if (t < 80) {
        float o;
        if (mode == 0) {
            o = A[t] * dv / (sqrtf(ss * (1.f / 79.f)) + 1e-6f) + Bp[t];
        } else {
            o = A[t] * dv / sqrtf(ss * (1.f / 80.f) + 1e-5f) + Bp[t];
            o = fmaxf(o, 0.f);
        }
        Y[(size_t)row * 80 + t] = o;
    }
}

// ---------------------------------------------------------------------------
// Zero-pad the layer-i FFN weights so DFF=120 becomes 128 (all GEMMs then have
// N,K multiples of 16/8 and need no masking).
//   W1p[80][128], b1p[128], W2p[128][80]
// ---------------------------------------------------------------------------
__global__ __launch_bounds__(256)
void pad_ffn_kernel(const float* __restrict__ W1, const float* __restrict__ b1,
                    const float* __restrict__ W2, float* __restrict__ W1p,
                    float* __restrict__ b1p, float* __restrict__ W2p)
{
    const int t = blockIdx.x * 256 + threadIdx.x;
    if (t < 80 * 128) {
        const int r = t >> 7, c = t & 127;
        W1p[t] = (c < 120) ? W1[r * 120 + c] : 0.f;
    } else if (t < 80 * 128 + 128) {
        const int c = t - 80 * 128;
        b1p[c] = (c < 120) ? b1[c] : 0.f;
    } else if (t < 80 * 128 + 128 + 128 * 80) {
        const int u = t - 80 * 128 - 128;
        const int r = u / 80, c = u - r * 80;
        W2p[u] = (r < 120) ? W2[r * 80 + c] : 0.f;
    }
}

// ---------------------------------------------------------------------------
// fp32 GEMM via V_WMMA_F32_16X16X4_F32.
// Block = 256 threads = 8 waves; block computes a 128(rows) x 16(cols) strip.
// B panel [K][16] is staged once in LDS (cooperative, conflict-free reads);
// A is read as even-aligned v2f (global_load_b64). K x2 unrolled.
// Requires: M % 128 == 0, N % 16 == 0, K % 8 == 0 (guaranteed by padding).
// TRANSB: B stored as [N][K] (used for sim = q @ M^T).
// ---------------------------------------------------------------------------
template <bool TRANSB>
__global__ __launch_bounds__(256)
void gemm_wmma_f32(const float* __restrict__ A, const float* __restrict__ B,
                   const float* __restrict__ bias, const float* __restrict__ R,
                   float* __restrict__ C, int N, int K, int relu)
{
    __shared__ float Bs[160 * 16];          // K <= 160
    const int n0 = blockIdx.x * 16;
    const int tid = threadIdx.x;
    for (int idx = tid; idx < K * 16; idx += 256) {
        const int kk = idx >> 4, cc = idx & 15;
        Bs[idx] = TRANSB ? B[(size_t)(n0 + cc) * K + kk]
                         : B[(size_t)kk * N + n0 + cc];
    }
    __syncthreads();

    const int wave = tid >> 5, lane = tid & 31;
    const int half = lane >> 4, l16 = lane & 15;
    const int m0 = (blockIdx.y * 8 + wave) * 16;
    const float* Arow = A + (size_t)(m0 + l16) * K + half * 2;

    v8f acc = {};
    for (int k = 0; k < K; k += 8) {
        const v2f a0 = *(const v2f*)(Arow + k);
        const v2f a1 = *(const v2f*)(Arow + k + 4);
        v2f b0, b1v;
        b0.x  = Bs[(k + half * 2 + 0) * 16 + l16];
        b0.y  = Bs[(k + half * 2 + 1) * 16 + l16];
        b1v.x = Bs[(k + 4 + half * 2 + 0) * 16 + l16];
        b1v.y = Bs[(k + 4 + half * 2 + 1) * 16 + l16];
        acc = __builtin_amdgcn_wmma_f32_16x16x4_f32(
            false, a0, false, b0, (short)0, acc, false, false);
        acc = __builtin_amdgcn_wmma_f32_16x16x4_f32(
            false, a1, false, b1v, (short)0, acc, false, false);
    }

    const int col = n0 + l16;
    const float bv = bias ? bias[col] : 0.f;
    for (int r = 0; r < 8; ++r) {
        const int row = m0 + half * 8 + r;
        float v = acc[r] + bv;
        if (R) v += R[(size_t)row * N + col];
        if (relu) v = fmaxf(v, 0.f);
        C[(size_t)row * N + col] = v;
    }
}

// ---------------------------------------------------------------------------
// Fused additive attention: one 256-thread block per (batch b, 16-row i-tile).
//   score[r,j] = sum_d tanh(Q[r,d]+K[j,d]) * c[d] + cb, c = wd+ws+wt + 1/sqrt(D)
//   softmax over j, then H[rows] += P[16x256] @ V[256x80] (WMMA, 5 waves).
// K tiles staged to LDS via async-load-to-LDS when available; V warmed via
// global_prefetch; P never leaves LDS.
// ---------------------------------------------------------------------------
__global__ __launch_bounds__(256)
void attn_kernel(const float* __restrict__ Q, const float* __restrict__ Kg,
                 const float* __restrict__ Vg, const float* __restrict__ wd,
                 const float* __restrict__ wsv, const float* __restrict__ wt,
                 const float* __restrict__ bd, const float* __restrict__ bsv,
                 const float* __restrict__ bt, float* __restrict__ H)
{
    const int b = blockIdx.x;
    const int i0 = blockIdx.y * 16;
    const int tid = threadIdx.x;

    __shared__ float Qs[16 * 80];
    __shared__ float Ks[16 * 84];   // stride 84 -> conflict-free strided reads
    __shared__ float S[16 * 260];   // stride 260 -> 32 distinct banks, 8B parity
    __shared__ float red[16 * 17];
    __shared__ float cvec[80];
    __shared__ float cb;

    if (tid < 80) cvec[tid] = wd[tid] + wsv[tid] + wt[tid] + 0.11180339887498948f;
    if (tid == 0) cb = bd[0] + bsv[0] + bt[0];
    const float* Qbase = Q + ((size_t)b * 256 + i0) * 80;
    for (int idx = tid; idx < 1280; idx += 256) Qs[idx] = Qbase[idx];

    // warm V panel for this batch into cache (global_prefetch_b8)
    const float* Vb = Vg + (size_t)b * 256 * 80;
    for (int off = tid * 64; off < 20480; off += 256 * 64)
        __builtin_prefetch(Vb + off, 0, 1);

    const int r = tid >> 4;    // output row 0..15
    const int jl = tid & 15;   // column within j-chunk

    for (int jt = 0; jt < 16; ++jt) {
        __syncthreads();  // previous chunk's Ks reads done (covers Qs/cvec on jt=0)
        const float* Kbase = Kg + ((size_t)b * 256 + jt * 16) * 80;
#ifdef USE_ASYNC_LDS
        for (int idx = tid; idx < 1280; idx += 256) {
            const int rr = idx / 80, dd = idx - rr * 80;
            __builtin_amdgcn_global_load_async_to_lds_b32(
                (async_gint*)(Kbase + idx),
                (async_lint*)(Ks + rr * 84 + dd),
                0, 0);
        }
        __builtin_amdgcn_s_wait_asynccnt(0);
#else
        for (int idx = tid; idx < 1280; idx += 256) {
            const int rr = idx / 80, dd = idx - rr * 80;
            Ks[rr * 84 + dd] = Kbase[idx];
        }
#endif
        __syncthreads();
        float s = 0.f;
        const float* qrow = Qs + r * 80;
        const float* krow = Ks + jl * 84;
        for (int d = 0; d < 80; ++d) s += tanhf(qrow[d] + krow[d]) * cvec[d];
        S[r * 260 + jt * 16 + jl] = s + cb;
    }
    __syncthreads();

    // ---- row softmax: 16 threads (jl) cooperate per row r ----
    float mx = -3.4e38f;
    for (int j = jl; j < 256; j += 16) mx = fmaxf(mx, S[r * 260 + j]);
    red[r * 17 + jl] = mx;
    __syncthreads();
    if (jl == 0) {
        float m = red[r * 17];
        for (int k = 1; k < 16; ++k) m = fmaxf(m, red[r * 17 + k]);
        red[r * 17] = m;
    }
    __syncthreads();
    mx = red[r * 17];
    __syncthreads();
    float sum = 0.f;
    for (int j = jl; j < 256; j += 16) {
        const float e = __expf(S[r * 260 + j] - mx);
        S[r * 260 + j] = e;
        sum += e;
    }
    red[r * 17 + jl] = sum;
    __syncthreads();
    if (jl == 0) {
        float m = 0.f;
        for (int k = 0; k < 16; ++k) m += red[r * 17 + k];
        red[r * 17] = 1.f / m;
    }
    __syncthreads();
    const float inv = red[r * 17];
    for (int j = jl; j < 256; j += 16) S[r * 260 + j] *= inv;
    __syncthreads();

    // ---- P @ V with f32 WMMA: waves 0..4 each own one 16-col tile of D=80 ----
    const int wave = tid >> 5;
    const int lane = tid & 31;
    if (wave < 5) {   // wave-uniform branch: EXEC stays all-ones for WMMA
        const int half = lane >> 4;
        const int l16 = lane & 15;
        const int col = wave * 16 + l16;
        v8f acc = {};
        for (int k = 0; k < 256; k += 4) {
            const v2f a = *(const v2f*)(S + l16 * 260 + k + half * 2); // ds_load_b64
            v2f bb;
            bb.x = Vb[(size_t)(k + half * 2 + 0) * 80 + col];
            bb.y = Vb[(size_t)(k + half * 2 + 1) * 80 + col];
            acc = __builtin_amdgcn_wmma_f32_16x16x4_f32(
                false, a, false, bb, (short)0, acc, false, false);
        }
        for (int rr = 0; rr < 8; ++rr) {  // residual add fused into store
            const size_t off = ((size_t)b * 256 + i0 + half * 8 + rr) * 80 + col;
            H[off] = H[off] + acc[rr];
        }
    }
}

// ---------------------------------------------------------------------------
// Memory module head: per row, top-3 of sim[64], 3-way softmax, weighted sum
// of memory rows; writes hcat[row] = [hn[row] (80), mem (80)].
// ---------------------------------------------------------------------------
__global__ __launch_bounds__(256)
void topk_mem_kernel(const float* __restrict__ sim, const float* __restrict__ Mm,
                     const float* __restrict__ hn, float* __restrict__ hcat,
                     int rows)
{
    const int row = blockIdx.x * 256 + threadIdx.x;
    if (row >= rows) return;
    const float* s = sim + (size_t)row * 64;
    float v0 = -3.4e38f, v1 = -3.4e38f, v2 = -3.4e38f;
    int i0 = 0, i1 = 0, i2 = 0;
    for (int j = 0; j < 64; ++j) {
        const float v = s[j];
        if (v > v0)      { v2 = v1; i2 = i1; v1 = v0; i1 = i0; v0 = v; i0 = j; }
        else if (v > v1) { v2 = v1; i2 = i1; v1 = v; i1 = j; }
        else if (v > v2) { v2 = v; i2 = j; }
    }
    float e0 = 1.f, e1 = __expf(v1 - v0), e2 = __expf(v2 - v0);
    const float inv = 1.f / (e0 + e1 + e2);
    e0 *= inv; e1 *= inv; e2 *= inv;
    float* o = hcat + (size_t)row * 160;
    const float* h = hn + (size_t)row * 80;
    const float* m0 = Mm + (size_t)i0 * 80;
    const float* m1 = Mm + (size_t)i1 * 80;
    const float* m2 = Mm + (size_t)i2 * 80;
    for (int d = 0; d < 80; ++d) {
        o[d] = h[d];
        o[80 + d] = e0 * m0[d] + e1 * m1[d] + e2 * m2[d];
    }
}

// ---------------------------------------------------------------------------
extern "C" void kernel_launch(void* const* d_in, const int* in_sizes, int n_in,
                              void* d_out, int out_size, void* d_ws, size_t ws_size,
                              hipStream_t stream)
{
    (void)in_sizes; (void)n_in; (void)out_size; (void)ws_size;
    const float* x    = (const float*)d_in[0];
    const float* Wq   = (const float*)d_in[1];
    const float* bq   = (const float*)d_in[2];
    const float* Wk   = (const float*)d_in[3];
    const float* bk   = (const float*)d_in[4];
    const float* Wv   = (const float*)d_in[5];
    const float* bv   = (const float*)d_in[6];
    const float* wd   = (const float*)d_in[7];
    const float* bd   = (const float*)d_in[8];
    const float* wsv  = (const float*)d_in[9];
    const float* bs   = (const float*)d_in[10];
    const float* wt   = (const float*)d_in[11];
    const float* bt   = (const float*)d_in[12];
    const float* W1   = (const float*)d_in[13];
    const float* b1   = (const float*)d_in[14];
    const float* W2   = (const float*)d_in[15];
    const float* b2   = (const float*)d_in[16];
    const float* ln_a = (const float*)d_in[17];
    const float* ln_b = (const float*)d_in[18];
    const float* na   = (const float*)d_in[19];
    const float* nb   = (const float*)d_in[20];
    const float* Mm   = (const float*)d_in[21];
    const float* Wqm  = (const float*)d_in[22];
    const float* bqm  = (const float*)d_in[23];
    const float* Wf   = (const float*)d_in[24];
    const float* bf   = (const float*)d_in[25];
    const float* fg   = (const float*)d_in[26];
    const float* fb   = (const float*)d_in[27];

    float* W = (float*)d_ws;
    const size_t Rr = 4096;               // B*T rows
    float* h    = W;                      // [4096,80]
    float* xn   = h    + Rr * 80;         // [4096,80]
    float* Qb   = xn   + Rr * 80;         // [4096,80]
    float* Kb   = Qb   + Rr * 80;         // [4096,80]
    float* Vb   = Kb   + Rr * 80;         // [4096,80]
    float* mid  = Vb   + Rr * 80;         // [4096,128]  (padded DFF)
    float* qm   = mid  + Rr * 128;        // [4096,80]
    float* sim  = qm   + Rr * 80;         // [4096,64]
    float* hcat = sim  + Rr * 64;         // [4096,160]
    float* hf   = hcat + Rr * 160;        // [4096,80]
    float* W1p  = hf   + Rr * 80;         // [80,128]
    float* b1p  = W1p  + 80 * 128;        // [128]
    float* W2p  = b1p  + 128;             // [128,80]

    (void)hipMemcpyAsync(h, x, Rr * 80 * sizeof(float), hipMemcpyDeviceToDevice, stream);

    const dim3 g80(5, 32), g128(8, 32), g64(4, 32);   // (N/16, M/128)
    for (int i = 0; i < 2; ++i) {
        ln80_kernel<<<4096, 128, 0, stream>>>(
            h, ln_a + (size_t)(i * 2 + 0) * 80, ln_b + (size_t)(i * 2 + 0) * 80, xn, 0);
        gemm_wmma_f32<false><<<g80, 256, 0, stream>>>(
            xn, Wq + (size_t)i * 80 * 80, bq + i * 80, nullptr, Qb, 80, 80, 0);
        gemm_wmma_f32<false><<<g80, 256, 0, stream>>>(
            xn, Wk + (size_t)i * 80 * 80, bk + i * 80, nullptr, Kb, 80, 80, 0);
        gemm_wmma_f32<false><<<g80, 256, 0, stream>>>(
            xn, Wv + (size_t)i * 80 * 80, bv + i * 80, nullptr, Vb, 80, 80, 0);
        attn_kernel<<<dim3(16, 16), 256, 0, stream>>>(Qb, Kb, Vb,
            wd + (size_t)i * 80, wsv + (size_t)i * 80, wt + (size_t)i * 80,
            bd + i, bs + i, bt + i, h);
        ln80_kernel<<<4096, 128, 0, stream>>>(
            h, ln_a + (size_t)(i * 2 + 1) * 80, ln_b + (size_t)(i * 2 + 1) * 80, xn, 0);
        pad_ffn_kernel<<<81, 256, 0, stream>>>(
            W1 + (size_t)i * 80 * 120, b1 + (size_t)i * 120, W2 + (size_t)i * 120 * 80,
            W1p, b1p, W2p);
        gemm_wmma_f32<false><<<g128, 256, 0, stream>>>(
            xn, W1p, b1p, nullptr, mid, 128, 80, 1);
        gemm_wmma_f32<false><<<g80, 256, 0, stream>>>(
            mid, W2p, b2 + i * 80, h, h, 80, 128, 0);
    }
    // final TCE norm + memory module
    ln80_kernel<<<4096, 128, 0, stream>>>(h, na, nb, xn, 0);
    gemm_wmma_f32<false><<<g80, 256, 0, stream>>>(xn, Wqm, bqm, nullptr, qm, 80, 80, 0);
    gemm_wmma_f32<true ><<<g64, 256, 0, stream>>>(qm, Mm, nullptr, nullptr, sim, 64, 80, 0);
    topk_mem_kernel<<<16, 256, 0, stream>>>(sim, Mm, xn, hcat, 4096);
    gemm_wmma_f32<false><<<g80, 256, 0, stream>>>(hcat, Wf, bf, nullptr, hf, 80, 160, 0);
    ln80_kernel<<<4096, 128, 0, stream>>>(hf, fg, fb, (float*)d_out, 1);
}